// ChebConv_31525059952899
// MI455X (gfx1250) — compile-verified
//
#include <hip/hip_runtime.h>

// Problem constants (from reference)
#define V_N     50000
#define B_N     8
#define FIN_N   64
#define FOUT_N  64
#define K_N     4
#define NNZ_N   800000
#define ROWLEN  512          // FIN_N * B_N  (dense feature row per vertex)
#define CDIM    256          // FIN_N * K_N  (GEMM reduction dim, c = f*4 + k)
#define LDS_PITCH 260        // 256 + 4 pad -> conflict-free b64 fragment reads

typedef __attribute__((ext_vector_type(2))) float v2f;
typedef __attribute__((ext_vector_type(8))) float v8f;

// ---------------------------------------------------------------------------
// 1) x[B,Fin,V] -> X0[v*512 + b*64 + f]   (32x32 LDS tile transpose)
//    x is [512, V] when flattened as c = b*64+f major: x[c*V + v]
// ---------------------------------------------------------------------------
__global__ void cheb_transpose(const float* __restrict__ x, float* __restrict__ X0) {
  __shared__ float t[32][33];
  const int tx = threadIdx.x, ty = threadIdx.y;
  const int vBase = blockIdx.x * 32, cBase = blockIdx.y * 32;
#pragma unroll
  for (int i = 0; i < 4; ++i) {
    const int c = cBase + ty + i * 8;
    const int v = vBase + tx;
    t[ty + i * 8][tx] = (v < V_N) ? x[(size_t)c * V_N + v] : 0.0f;
  }
  __syncthreads();
#pragma unroll
  for (int i = 0; i < 4; ++i) {
    const int v = vBase + ty + i * 8;
    const int c = cBase + tx;
    if (v < V_N) X0[(size_t)v * ROWLEN + c] = t[tx][ty + i * 8];
  }
}

// ---------------------------------------------------------------------------
// 2) CSR build: histogram -> single-block scan -> scatter
// ---------------------------------------------------------------------------
__global__ void cheb_hist(const int* __restrict__ rows, int* __restrict__ cnt) {
  const int e = blockIdx.x * blockDim.x + threadIdx.x;
  if (e < NNZ_N) atomicAdd(&cnt[rows[e]], 1);
}

__global__ void cheb_scan(const int* __restrict__ cnt, int* __restrict__ rp,
                          int* __restrict__ ctr) {
  __shared__ int sums[1024];
  const int CH = (V_N + 1023) / 1024;  // 49
  const int t = threadIdx.x;
  const int beg = t * CH;
  const int end = (beg + CH < V_N) ? (beg + CH) : V_N;
  int s = 0;
  for (int i = beg; i < end; ++i) s += cnt[i];
  sums[t] = s;
  __syncthreads();
  // Hillis-Steele inclusive scan over 1024 thread-sums
  for (int off = 1; off < 1024; off <<= 1) {
    const int val = (t >= off) ? sums[t - off] : 0;
    __syncthreads();
    sums[t] += val;
    __syncthreads();
  }
  int run = (t == 0) ? 0 : sums[t - 1];
  for (int i = beg; i < end; ++i) {
    rp[i] = run;
    ctr[i] = run;      // scatter cursor starts at row begin
    run += cnt[i];
  }
  if (t == 1023) rp[V_N] = sums[1023];
}

__global__ void cheb_scatter(const int* __restrict__ rows, const int* __restrict__ cols,
                             const float* __restrict__ vals, int* __restrict__ ctr,
                             int* __restrict__ ccol, float* __restrict__ cval) {
  const int e = blockIdx.x * blockDim.x + threadIdx.x;
  if (e < NNZ_N) {
    const int r = rows[e];
    const int p = atomicAdd(&ctr[r], 1);
    ccol[p] = cols[e];
    cval[p] = vals[e];
  }
}

// ---------------------------------------------------------------------------
// 3) Row-parallel CSR SpMM + Chebyshev combine.
//    cheb==0: Xout = L @ Xin ;  cheb==1: Xout = 2*(L @ Xin) - Xprev
//    One 128-thread block per vertex row; each thread owns 4 floats of the
//    512-float row -> fully coalesced 2KB row gathers (L2-resident operand).
// ---------------------------------------------------------------------------
__global__ void __launch_bounds__(128)
cheb_spmm(const int* __restrict__ rp, const int* __restrict__ ccol,
          const float* __restrict__ cval, const float* __restrict__ Xin,
          const float* __restrict__ Xprev, float* __restrict__ Xout, int cheb) {
  const int row = blockIdx.x;
  const int base = threadIdx.x * 4;
  float4 acc = make_float4(0.f, 0.f, 0.f, 0.f);
  const int e0 = rp[row], e1 = rp[row + 1];
  for (int e = e0; e < e1; ++e) {
    const int c = ccol[e];
    const float w = cval[e];
    const float4 xv = *reinterpret_cast<const float4*>(Xin + (size_t)c * ROWLEN + base);
    acc.x += w * xv.x; acc.y += w * xv.y; acc.z += w * xv.z; acc.w += w * xv.w;
  }
  if (cheb) {
    const float4 pv = *reinterpret_cast<const float4*>(Xprev + (size_t)row * ROWLEN + base);
    acc.x = 2.f * acc.x - pv.x; acc.y = 2.f * acc.y - pv.y;
    acc.z = 2.f * acc.z - pv.z; acc.w = 2.f * acc.w - pv.w;
  }
  *reinterpret_cast<float4*>(Xout + (size_t)row * ROWLEN + base) = acc;
}

// ---------------------------------------------------------------------------
// 4) WMMA GEMM: out[b,fo,v] = sum_c W[fo,c] * Xc[(b,v), c],  c = f*4 + k.
//    A = W tile (16 fo x 4 c), B = X tile (4 c x 16 v) -> D(N=v across lanes)
//    gives coalesced 64B output stores per accumulator VGPR.
//    Block = 128 threads (4 waves, one fout-tile each) sharing one LDS B-tile (X).
// ---------------------------------------------------------------------------
__global__ void __launch_bounds__(128)
cheb_gemm_wmma(const float* __restrict__ Xbase,   // X0; X_k at k*V_N*ROWLEN
               const float* __restrict__ Wm,      // [64, 256], col c = f*4+k
               float* __restrict__ out) {         // [B, FOUT, V]
  __shared__ float lds[16 * LDS_PITCH];
  const int tid = threadIdx.x;
  const int row0 = blockIdx.x * 16;         // global (b,v) row tile
  const int b = row0 / V_N;                 // V_N % 16 == 0 -> single b per tile
  const int v0 = row0 - b * V_N;

  // Stage X tile into LDS as [r][f*4+k] with pitch 260.
  // idx -> (f = idx&63, k = (idx>>6)&3, r = idx>>8): 64 consecutive threads
  // read one contiguous 256B segment of X_k -> coalesced.
#pragma unroll
  for (int i = 0; i < 32; ++i) {
    const int idx = tid + i * 128;          // 0..4095
    const int f = idx & 63;
    const int k = (idx >> 6) & 3;
    const int r = idx >> 8;
    lds[r * LDS_PITCH + f * 4 + k] =
        Xbase[(size_t)k * V_N * ROWLEN + (size_t)(v0 + r) * ROWLEN + b * FIN_N + f];
  }
  __syncthreads();

  const int wave = tid >> 5;
  const int lane = tid & 31;
  const int l16  = lane & 15;
  const int half = lane >> 4;
  const int fo0  = wave * 16;

  // Fragment base pointers (8B-aligned for float2 / v2f reads):
  // A (16x4 f32): lane<16 holds M=l16, K=c0..c0+1; lane>=16 holds K=c0+2..c0+3
  // B (4x16 f32): lane<16 holds N=l16, K=c0..c0+1; lane>=16 holds K=c0+2..c0+3
  const float* wrow = Wm + (size_t)(fo0 + l16) * CDIM + 2 * half;
  const float* xrow = lds + l16 * LDS_PITCH + 2 * half;

  v8f acc = {};
#pragma unroll 8
  for (int c0 = 0; c0 < CDIM; c0 += 4) {
    const v2f a  = *reinterpret_cast<const v2f*>(wrow + c0);
    const v2f bb = *reinterpret_cast<const v2f*>(xrow + c0);
    acc = __builtin_amdgcn_wmma_f32_16x16x4_f32(
        /*neg_a=*/false, a, /*neg_b=*/false, bb,
        /*c_mod=*/(short)0, acc, /*reuse_a=*/false, /*reuse_b=*/false);
  }

  // D layout: VGPR j, lanes 0-15 -> (M=j, N=lane); lanes 16-31 -> (M=j+8, N=lane-16)
  // M = fout within tile, N = v within tile -> 64B-contiguous stores per j.
  const size_t obase = (size_t)b * FOUT_N * V_N + (size_t)v0 + l16;
#pragma unroll
  for (int j = 0; j < 8; ++j) {
    const int fo = fo0 + j + 8 * half;
    out[obase + (size_t)fo * V_N] = acc[j];
  }
}

// ---------------------------------------------------------------------------
extern "C" void kernel_launch(void* const* d_in, const int* in_sizes, int n_in,
                              void* d_out, int out_size, void* d_ws, size_t ws_size,
                              hipStream_t stream) {
  const float* x     = (const float*)d_in[0];   // [B, FIN, V]
  const float* Lvals = (const float*)d_in[1];   // [NNZ]
  const float* Wm    = (const float*)d_in[2];   // [FOUT, K*FIN]
  const int*   Lrows = (const int*)d_in[3];     // [NNZ]
  const int*   Lcols = (const int*)d_in[4];     // [NNZ]
  float* out = (float*)d_out;                   // [B, FOUT, V]

  // Workspace layout (~417 MB total)
  char* ws = (char*)d_ws;
  const size_t xkElems = (size_t)V_N * ROWLEN;            // 25.6M floats each
  float* X0 = (float*)ws;
  float* X1 = X0 + xkElems;
  float* X2 = X1 + xkElems;
  float* X3 = X2 + xkElems;
  int*   cnt  = (int*)(ws + 4 * xkElems * sizeof(float));
  int*   rp   = cnt + V_N;           // V_N + 1 entries
  int*   ctr  = rp + (V_N + 1);
  int*   ccol = ctr + V_N;           // NNZ entries
  float* cval = (float*)(ccol + NNZ_N);

  // CSR build
  hipMemsetAsync(cnt, 0, V_N * sizeof(int), stream);
  cheb_transpose<<<dim3((V_N + 31) / 32, ROWLEN / 32), dim3(32, 8), 0, stream>>>(x, X0);
  cheb_hist<<<(NNZ_N + 255) / 256, 256, 0, stream>>>(Lrows, cnt);
  cheb_scan<<<1, 1024, 0, stream>>>(cnt, rp, ctr);
  cheb_scatter<<<(NNZ_N + 255) / 256, 256, 0, stream>>>(Lrows, Lcols, Lvals, ctr, ccol, cval);

  // Chebyshev recurrence (3 SpMM passes, L2-resident dense operand)
  cheb_spmm<<<V_N, 128, 0, stream>>>(rp, ccol, cval, X0, X0, X1, 0);
  cheb_spmm<<<V_N, 128, 0, stream>>>(rp, ccol, cval, X1, X0, X2, 1);
  cheb_spmm<<<V_N, 128, 0, stream>>>(rp, ccol, cval, X2, X1, X3, 1);

  // Fused WMMA GEMM + output transpose
  cheb_gemm_wmma<<<(B_N * V_N) / 16, 128, 0, stream>>>(X0, Wm, out);
}